// SAGE_24515673325905
// MI455X (gfx1250) — compile-verified
//
#include <hip/hip_runtime.h>

typedef float v2f __attribute__((ext_vector_type(2)));
typedef float v8f __attribute__((ext_vector_type(8)));

#define F_DIM 64          // feature dim (both in and hidden)
#define K_DIM 128         // concat dim
#define PAIRS (K_DIM / 2) // 64 even/odd k-row pairs
#define PSTR  130         // dwords per pair-row: 64 cols * 2 + 2 pad (keeps 8B align)

// ---------------------------------------------------------------------------
// Edge-degree count: cnt[dst] += 1
// ---------------------------------------------------------------------------
__global__ __launch_bounds__(256) void k_count(const long long* __restrict__ ei,
                                               float* __restrict__ cnt, int E) {
    int e = blockIdx.x * 256 + threadIdx.x;
    if (e < E) {
        int d = (int)ei[(size_t)E + e];   // row 1 = dst
        unsafeAtomicAdd(&cnt[d], 1.0f);
    }
}

// cnt[i] = 1 / max(cnt[i], 1)
__global__ __launch_bounds__(256) void k_recip(float* __restrict__ cnt, int n) {
    int i = blockIdx.x * 256 + threadIdx.x;
    if (i < n) cnt[i] = 1.0f / fmaxf(cnt[i], 1.0f);
}

// ---------------------------------------------------------------------------
// Scatter-add of neighbor features: agg[dst][*] += F[src][*]
// 16 threads per edge, float4 per thread (coalesced 256B gather per edge row)
// ---------------------------------------------------------------------------
__global__ __launch_bounds__(256) void k_scatter(const float* __restrict__ F,
                                                 const long long* __restrict__ ei,
                                                 float* __restrict__ agg, int E) {
    long long t = (long long)blockIdx.x * 256 + threadIdx.x;
    int e = (int)(t >> 4);
    if (e >= E) return;
    int c = (int)(t & 15) << 2;                       // feature offset 0..60
    long long s = ei[e];                              // row 0 = src
    long long d = ei[(size_t)E + e];                  // row 1 = dst
    const float4 v = *(const float4*)(F + s * F_DIM + c);
    float* p = agg + d * F_DIM + c;
    unsafeAtomicAdd(p + 0, v.x);
    unsafeAtomicAdd(p + 1, v.y);
    unsafeAtomicAdd(p + 2, v.z);
    unsafeAtomicAdd(p + 3, v.w);
}

// ---------------------------------------------------------------------------
// Fused SAGE linear:  OUT = act( [X || AGG*inv] @ W + b )
// W staged in LDS k-pair-interleaved: sW[k/2][col][k&1], so each 16x16x4 B
// fragment {W[k][col], W[k+1][col]} is one aligned ds_load_b64 straight into
// the WMMA source register pair (no register shuffles).
// One wave = 16 rows x 64 cols strip = 32 K-steps x 4 tiles of
// V_WMMA_F32_16X16X4_F32, fully unrolled and branch-free.
// Block = 256 threads = 8 waves = 128 rows.
// ---------------------------------------------------------------------------
__global__ __launch_bounds__(256) void sage_gemm(const float* __restrict__ X,
                                                 const float* __restrict__ AGG,
                                                 const float* __restrict__ INV,
                                                 const float* __restrict__ W,
                                                 const float* __restrict__ B,
                                                 float* __restrict__ OUT,
                                                 int nrows, int relu) {
    __shared__ float sW[PAIRS * PSTR];
    for (int i = threadIdx.x; i < K_DIM * F_DIM; i += 256) {
        int r = i >> 6, c = i & 63;                    // W[r][c], row-major source
        sW[(r >> 1) * PSTR + (c << 1) + (r & 1)] = W[i];
    }
    __syncthreads();

    const int lane  = threadIdx.x & 31;
    const int wave  = threadIdx.x >> 5;
    const int mrow  = lane & 15;          // M index (A) / N index (B,C,D) within tile
    const int khalf = lane >> 4;          // 0 for lanes 0-15, 1 for lanes 16-31

    const int rowBase = blockIdx.x * 128 + wave * 16;
    int arow = rowBase + mrow;
    int arowC = arow < nrows ? arow : (nrows - 1);    // clamp; stores are masked
    const float inv = INV[arowC];
    const float* __restrict__ xrow = X   + (size_t)arowC * F_DIM;
    const float* __restrict__ grow = AGG + (size_t)arowC * F_DIM;

    // bias-initialized accumulators: C element depends only on global column
    v8f acc[4];
#pragma unroll
    for (int nt = 0; nt < 4; ++nt) {
        float bv = B[nt * 16 + mrow];
#pragma unroll
        for (int r = 0; r < 8; ++r) acc[nt][r] = bv;
    }

    // ---- phase 1: K = 0..63, self features ----
#pragma unroll
    for (int ks = 0; ks < 16; ++ks) {
        const int k0 = ks * 4 + khalf * 2;            // even
        v2f a;
        a.x = xrow[k0];
        a.y = xrow[k0 + 1];
        const int kp = ks * 2 + khalf;                // k-pair index = k0/2
#pragma unroll
        for (int nt = 0; nt < 4; ++nt) {
            v2f b = *(const v2f*)&sW[kp * PSTR + ((nt * 16 + mrow) << 1)];
            acc[nt] = __builtin_amdgcn_wmma_f32_16x16x4_f32(
                false, a, false, b, (short)0, acc[nt], false, false);
        }
    }

    // ---- phase 2: K = 64..127, mean-aggregated neighbor features ----
#pragma unroll
    for (int ks = 0; ks < 16; ++ks) {
        const int k0 = ks * 4 + khalf * 2;
        v2f a;
        a.x = grow[k0] * inv;
        a.y = grow[k0 + 1] * inv;
        const int kp = 32 + ks * 2 + khalf;           // second half of k-pairs
#pragma unroll
        for (int nt = 0; nt < 4; ++nt) {
            v2f b = *(const v2f*)&sW[kp * PSTR + ((nt * 16 + mrow) << 1)];
            acc[nt] = __builtin_amdgcn_wmma_f32_16x16x4_f32(
                false, a, false, b, (short)0, acc[nt], false, false);
        }
    }

    // C/D layout: VGPR r -> M = r + 8*(lane>=16), N = lane&15
    const int mhi = khalf << 3;
#pragma unroll
    for (int nt = 0; nt < 4; ++nt) {
#pragma unroll
        for (int r = 0; r < 8; ++r) {
            int row = rowBase + r + mhi;
            if (row < nrows) {
                float v = acc[nt][r];
                if (relu) v = fmaxf(v, 0.0f);
                OUT[(size_t)row * F_DIM + nt * 16 + mrow] = v;
            }
        }
    }
}

// ---------------------------------------------------------------------------
extern "C" void kernel_launch(void* const* d_in, const int* in_sizes, int n_in,
                              void* d_out, int out_size, void* d_ws, size_t ws_size,
                              hipStream_t stream) {
    const float*     x  = (const float*)d_in[0];
    const long long* ei = (const long long*)d_in[1];  // int64 [2,E]
    const float*     W1 = (const float*)d_in[2];
    const float*     b1 = (const float*)d_in[3];
    const float*     W2 = (const float*)d_in[4];
    const float*     b2 = (const float*)d_in[5];
    float*           out = (float*)d_out;

    const int N = in_sizes[0] / F_DIM;   // 50000
    const int E = in_sizes[1] / 2;       // 800000

    float* cnt = (float*)d_ws;                  // [N]  (becomes 1/deg)
    float* agg = cnt + N;                       // [N,64] (reused both layers)
    float* h1  = agg + (size_t)N * F_DIM;       // [N,64]

    const int eb  = (E + 255) / 256;
    const int sb  = (int)(((long long)E * 16 + 255) / 256);
    const int nb  = (N + 255) / 256;
    const int gb  = (N + 127) / 128;

    // zero cnt + agg (contiguous) for the atomic accumulation
    hipMemsetAsync(cnt, 0, sizeof(float) * (size_t)N * (1 + F_DIM), stream);

    // --- layer 1 ---
    k_count  <<<eb, 256, 0, stream>>>(ei, cnt, E);
    k_scatter<<<sb, 256, 0, stream>>>(x, ei, agg, E);
    k_recip  <<<nb, 256, 0, stream>>>(cnt, N);
    sage_gemm<<<gb, 256, 0, stream>>>(x, agg, cnt, W1, b1, h1, N, 1);

    // --- layer 2 ---
    hipMemsetAsync(agg, 0, sizeof(float) * (size_t)N * F_DIM, stream);
    k_scatter<<<sb, 256, 0, stream>>>(h1, ei, agg, E);
    sage_gemm<<<gb, 256, 0, stream>>>(h1, agg, cnt, W2, b2, out, N, 0);
}